// Distance_64166811402487
// MI455X (gfx1250) — compile-verified
//
#include <hip/hip_runtime.h>
#include <hip/hip_bf16.h>

typedef __attribute__((ext_vector_type(2))) float v2f;
typedef __attribute__((ext_vector_type(8))) float v8f;
typedef __attribute__((ext_vector_type(4))) float f32x4;

constexpr int   BB = 32;
constexpr int   NN = 2048;
constexpr int   DD = 64;
constexpr float MAX_DIST_SQ = 0.25f;   // 0.5^2

// ---------------------------------------------------------------------------
// Kernel 1: per-(b,n) mask = ||nodes[b,n] - nodes[b,nn[b]]||^2 < 0.25
// One wave handles a 16-row tile. WMMA Gram trick: with the CDNA5 f32 16x16x4
// layouts, passing the same lane registers as A and B computes D = diff*diffT;
// the diagonal of D accumulated over 16 K-chunks is the squared distance.
// ---------------------------------------------------------------------------
__global__ void __launch_bounds__(256)
mask_kernel(const float* __restrict__ nodes, const int* __restrict__ num_nodes,
            unsigned char* __restrict__ mask) {
    const int wave = (blockIdx.x * blockDim.x + threadIdx.x) >> 5;
    const int lane = threadIdx.x & 31;
    const int tilesPerBatch = NN / 16;               // 128
    const int b    = wave / tilesPerBatch;
    const int tile = wave % tilesPerBatch;
    const int h    = lane >> 4;                      // half-wave selects K pair
    const int row  = tile * 16 + (lane & 15);

    const int cur = num_nodes[b];
    const float* __restrict__ xr = nodes + ((size_t)b * NN + row) * DD;
    const float* __restrict__ cr = nodes + ((size_t)b * NN + cur) * DD;

    v8f acc = {};
#pragma unroll
    for (int chunk = 0; chunk < 16; ++chunk) {
        const int k = chunk * 4 + 2 * h;             // A layout: lanes0-15 K={0,1}, lanes16-31 K={2,3}
        v2f a;
        a.x = xr[k]     - cr[k];
        a.y = xr[k + 1] - cr[k + 1];
        // SRC1 == SRC0 realizes B = A^T under the CDNA5 f32 B-matrix layout.
        acc = __builtin_amdgcn_wmma_f32_16x16x4_f32(
            /*neg_a=*/false, a, /*neg_b=*/false, a,
            /*c_mod=*/(short)0, acc, /*reuse_a=*/false, /*reuse_b=*/false);
    }

    // Diagonal D[i][i]: i<8 -> (vgpr i, lane i); i>=8 -> (vgpr i-8, lane i+16).
    int sel = -1, orow = 0;
    if (lane < 8)        { sel = lane;      orow = lane;      }
    else if (lane >= 24) { sel = lane - 24; orow = lane - 16; }
    if (sel >= 0) {
        float d2;
        switch (sel) {
            case 0: d2 = acc[0]; break;
            case 1: d2 = acc[1]; break;
            case 2: d2 = acc[2]; break;
            case 3: d2 = acc[3]; break;
            case 4: d2 = acc[4]; break;
            case 5: d2 = acc[5]; break;
            case 6: d2 = acc[6]; break;
            default: d2 = acc[7]; break;
        }
        mask[(size_t)b * NN + tile * 16 + orow] = (d2 < MAX_DIST_SQ) ? 1u : 0u;
    }
}

// ---------------------------------------------------------------------------
// Kernel 2: the bandwidth-bound part. Stream adj_mats and edge_weights to the
// output with 16B vector non-temporal loads/stores (1 GiB in + 1 GiB out;
// nothing is reused, so bypass cache retention).
// ---------------------------------------------------------------------------
__global__ void __launch_bounds__(256)
copy_kernel(const f32x4* __restrict__ adj_in, const f32x4* __restrict__ ew_in,
            f32x4* __restrict__ adj_out, f32x4* __restrict__ ew_out, size_t n4) {
    size_t i = (size_t)blockIdx.x * blockDim.x + threadIdx.x;
    const size_t stride = (size_t)gridDim.x * blockDim.x;
    for (; i < n4; i += stride) {
        f32x4 va = __builtin_nontemporal_load(&adj_in[i]);
        f32x4 vw = __builtin_nontemporal_load(&ew_in[i]);
        __builtin_nontemporal_store(va, &adj_out[i]);
        __builtin_nontemporal_store(vw, &ew_out[i]);
    }
}

// ---------------------------------------------------------------------------
// Kernel 3: tiny scatter fixup. Where mask[b,n]: adj[b][nn][n]=1, adj[b][n][nn]=1.
// (At n==nn both writes hit the same cell with mask true -> matches reference.)
// ---------------------------------------------------------------------------
__global__ void __launch_bounds__(256)
fixup_kernel(const unsigned char* __restrict__ mask, const int* __restrict__ num_nodes,
             float* __restrict__ adj) {
    const int t = blockIdx.x * blockDim.x + threadIdx.x;   // 0 .. B*N-1
    const int b = t >> 11;                                 // /N
    const int n = t & (NN - 1);
    if (mask[t]) {
        const int cur = num_nodes[b];
        float* base = adj + (size_t)b * NN * NN;
        base[(size_t)cur * NN + n] = 1.0f;   // row scatter (coalesced)
        base[(size_t)n * NN + cur] = 1.0f;   // column scatter (strided)
    }
}

extern "C" void kernel_launch(void* const* d_in, const int* in_sizes, int n_in,
                              void* d_out, int out_size, void* d_ws, size_t ws_size,
                              hipStream_t stream) {
    const float* nodes     = (const float*)d_in[0];
    const float* adj_mats  = (const float*)d_in[1];
    const float* edge_wts  = (const float*)d_in[2];
    const int*   num_nodes = (const int*)d_in[3];
    (void)in_sizes; (void)n_in; (void)out_size; (void)ws_size;

    float* out_adj = (float*)d_out;
    float* out_ew  = out_adj + (size_t)BB * NN * NN;
    unsigned char* mask = (unsigned char*)d_ws;            // B*N = 64 KiB

    // 1) mask via WMMA: 32*128 = 4096 waves, 8 waves/block -> 512 blocks
    mask_kernel<<<512, 256, 0, stream>>>(nodes, num_nodes, mask);

    // 2) streaming copy: 32*2048*2048/4 float4 per array
    const size_t n4 = ((size_t)BB * NN * NN) / 4;          // 33,554,432
    copy_kernel<<<16384, 256, 0, stream>>>((const f32x4*)adj_mats, (const f32x4*)edge_wts,
                                           (f32x4*)out_adj, (f32x4*)out_ew, n4);

    // 3) fixup scatter: B*N threads
    fixup_kernel<<<(BB * NN) / 256, 256, 0, stream>>>(mask, num_nodes, out_adj);
}